// LocalHistogramLayer1_40175124087485
// MI455X (gfx1250) — compile-verified
//
#include <hip/hip_runtime.h>

typedef __attribute__((ext_vector_type(2))) float v2f;
typedef __attribute__((ext_vector_type(8))) float v8f;

#define HW   65536
#define CIN  8
#define COUT 16
#define LOG2E 1.44269504088896340736f

// One wave computes a 16-pixel x 16-channel output tile:
//   hist_tile[16x16] = E[16x128] * S[128x16]
// where E are the exp2 terms and S is the block-diagonal j->o selection matrix,
// evaluated as 32 chained V_WMMA_F32_16X16X4_F32 (K=4 each).
__global__ __launch_bounds__(256)
void rbf_hist_wmma_kernel(const float* __restrict__ x,
                          const float* __restrict__ centers,
                          const float* __restrict__ widths,
                          float* __restrict__ out)
{
    // 64 oj-pairs, each packed {c(2p), c(2p+1), -w'(2p), -w'(2p+1)} so a single
    // b128 load yields pk-ready adjacent pairs (centers / pre-negated scales).
    __shared__ float sParams[256];
    __shared__ float sTile[8][320];    // per-wave 16 rows x stride 20 (bank-conflict free, 16B aligned)

    const int tid  = threadIdx.x;
    const int lane = tid & 31;
    const int wave = tid >> 5;
    const int hi   = (lane >> 4) & 1;  // upper half-wave
    const int n    = lane & 15;        // pixel row for A; channel column for B/D

    if (tid < 128) {
        float c   = centers[tid];
        float w   = widths[tid];
        float inv = 1.0f / (2.0f * w * w);
        const int pr = tid >> 1;       // pair index
        const int e  = tid & 1;
        sParams[pr * 4 + e]     = c;
        sParams[pr * 4 + 2 + e] = -(inv * LOG2E);   // pre-negated: exp2(d*d*w') directly
    }
    __syncthreads();

    // tile -> (batch, pixel base). 4096 tiles per batch image, 8 batches, exact coverage.
    const int tile = blockIdx.x * 8 + wave;
    const int b    = tile >> 12;
    const int hw   = (tile & 4095) << 4;

    // Each lane only ever needs j in {2hi, 2hi+1, 4+2hi, 4+2hi+1}: fold hi into the
    // base address and load exactly 4 floats per lane (no in-loop register selection,
    // and no duplicate loads between half-waves: 32 lanes x 4 = 16px x 8j exactly).
    const float* xp = x + (size_t)b * CIN * HW + hw + n + (size_t)(hi ? 2 * HW : 0);
    const float xe0 = xp[0];                    // j = 2hi      (even steps)
    const float xe1 = xp[HW];                   // j = 2hi + 1
    const float xo0 = xp[4 * (size_t)HW];       // j = 4 + 2hi  (odd steps)
    const float xo1 = xp[5 * (size_t)HW];       // j = 5 + 2hi

    const float4* p4 = (const float4*)sParams;
    v8f acc = {0.f, 0.f, 0.f, 0.f, 0.f, 0.f, 0.f, 0.f};

#pragma unroll
    for (int s = 0; s < 32; ++s) {
        // Step s covers oj = 4s..4s+3; lower half-wave supplies K=0,1 (oj=4s,4s+1),
        // upper half K=2,3 (oj=4s+2,4s+3). o = s>>1 for all four.
        float4 pw = p4[2 * s + hi];             // {c0, c1, -w0', -w1'}
        float u0 = (s & 1) ? xo0 : xe0;         // compile-time select after unroll
        float u1 = (s & 1) ? xo1 : xe1;
        float d0 = u0 - pw.x;
        float d1 = u1 - pw.y;
        v2f a;
        a[0] = __builtin_amdgcn_exp2f(d0 * d0 * pw.z);   // v_exp_f32, scale already negated
        a[1] = __builtin_amdgcn_exp2f(d1 * d1 * pw.w);
        // Selection matrix column: all 4 K-rows equal (o == n) ? 1 : 0 this step.
        float bb = (n == (s >> 1)) ? 1.0f : 0.0f;
        v2f bv; bv[0] = bb; bv[1] = bb;
        acc = __builtin_amdgcn_wmma_f32_16x16x4_f32(false, a, false, bv,
                                                    (short)0, acc, false, false);
    }

    // D layout: acc[r] holds (pixel = r + hi*8, channel = n).
    // Transpose through padded LDS so global stores are contiguous per channel.
    float* myTile = &sTile[wave][0];
    const int rowbase = hi << 3;
#pragma unroll
    for (int r = 0; r < 8; ++r)
        myTile[n * 20 + rowbase + r] = acc[r];

    __builtin_amdgcn_wave_barrier();   // same-wave DS ops are in-order; fence the scheduler

    const int o2 = lane >> 1;          // channel this lane stores
    const int ph = (lane & 1) << 3;    // pixel half (0 or 8)
    const float4* t4 = (const float4*)(myTile + o2 * 20 + ph);
    float4 q0 = t4[0];
    float4 q1 = t4[1];

    float* op = out + ((size_t)b * COUT + o2) * HW + hw + ph;
    ((float4*)op)[0] = q0;             // 64B contiguous per lane-pair -> coalesced b128 stores
    ((float4*)op)[1] = q1;
}

extern "C" void kernel_launch(void* const* d_in, const int* in_sizes, int n_in,
                              void* d_out, int out_size, void* d_ws, size_t ws_size,
                              hipStream_t stream) {
    const float* x       = (const float*)d_in[0];
    const float* centers = (const float*)d_in[1];
    const float* widths  = (const float*)d_in[2];
    float* out = (float*)d_out;

    // 32768 tiles total = 4096 blocks x 8 waves (exact, no tail -> EXEC all-ones for WMMA)
    rbf_hist_wmma_kernel<<<dim3(4096), dim3(256), 0, stream>>>(x, centers, widths, out);
}